// DiffusionTimePredictor_89970974917004
// MI455X (gfx1250) — compile-verified
//
#include <hip/hip_runtime.h>
#include <math.h>

typedef float v2f __attribute__((ext_vector_type(2)));
typedef float v8f __attribute__((ext_vector_type(8)));

#define NHEAD 16
#define SQ 4096
#define SK 4096
#define DK 64
#define QTILES (SQ / 16)          // 256 q-tiles per head
#define KTILES (SK / 16)          // 256 k-tiles
#define WAVES 8
#define KT_PER_WAVE (KTILES / WAVES)  // 32
#define KT_STRIDE ((size_t)WAVES * 16 * DK)   // floats between consecutive tiles of a wave

// sum/max across the 16-lane halves of a wave32 (masks 1,2,4,8 stay within each half)
__device__ __forceinline__ float xor16_sum(float v) {
#pragma unroll
  for (int off = 1; off < 16; off <<= 1) v += __shfl_xor(v, off, 32);
  return v;
}
__device__ __forceinline__ float xor16_max(float v) {
#pragma unroll
  for (int off = 1; off < 16; off <<= 1) v = fmaxf(v, __shfl_xor(v, off, 32));
  return v;
}

__device__ __forceinline__ void load_ktile(v2f* kb, const float* Kp) {
#pragma unroll
  for (int t = 0; t < 16; ++t) kb[t] = *(const v2f*)(Kp + 4 * t);
}

__device__ __forceinline__ void consume_tile(const v2f* qa, const v2f* kb,
                                             float* accS, float* accM,
                                             float* accZ, float* accS2) {
  v8f c = {0.f, 0.f, 0.f, 0.f, 0.f, 0.f, 0.f, 0.f};
#pragma unroll
  for (int t = 0; t < 16; ++t)
    c = __builtin_amdgcn_wmma_f32_16x16x4_f32(false, qa[t], false, kb[t],
                                              (short)0, c, false, false);
#pragma unroll
  for (int j = 0; j < 8; ++j) {
    float s = c[j] * 0.125f;                    // / sqrt(64)
    s = fminf(fmaxf(s, -50.f), 50.f);           // clip; makes exp() overflow-free
    accS[j] += s;
    accM[j] = fmaxf(accM[j], s);
    float e = __expf(s * 0.5f);                 // softmax(s/2) term; no max-subtraction needed
    accZ[j] += e;
    accS2[j] = fmaf(e, e, accS2[j]);            // e^s
  }
}

// One block = one (head, q-tile). Produces 3 partial sums per block:
//   [0] sum over (16 rows x 4096 cols) of clipped score
//   [1] sum over 16 rows of rowwise max
//   [2] sum over 16 rows of rowwise unbiased variance of softmax(s/2)
__global__ __launch_bounds__(256)
void attn_stats_kernel(const float* __restrict__ Q,
                       const float* __restrict__ K,
                       float* __restrict__ partials) {
  const int bid  = blockIdx.x;
  const int h    = bid >> 8;     // head
  const int qt   = bid & 255;    // q tile
  const int tid  = threadIdx.x;
  const int wave = tid >> 5;
  const int lane = tid & 31;
  const int m    = lane & 15;
  const int hh   = lane >> 4;    // half: selects K=0,1 vs K=2,3 per f32 WMMA A/B layout

  __shared__ float wstats[WAVES][16][4];
  __shared__ float rowtrip[16][3];

  // A-matrix (Q tile, 16x4 per step): lane (m,hh) holds Q[qrow=m][4t + 2*hh + {0,1}]
  const float* Qp = Q + (((size_t)h * SQ + (size_t)qt * 16 + m) * DK) + 2 * hh;
  v2f qa[16];
#pragma unroll
  for (int t = 0; t < 16; ++t) qa[t] = *(const v2f*)(Qp + 4 * t);

  // per-lane accumulators for the 8 C-slots (rows j / j+8, col class = lane&15)
  float accS[8], accM[8], accZ[8], accS2[8];
#pragma unroll
  for (int j = 0; j < 8; ++j) {
    accS[j] = 0.f; accM[j] = -1e30f; accZ[j] = 0.f; accS2[j] = 0.f;
  }

  // B-matrix base for this wave's first k-tile (kt = wave); stride 8 tiles between iters
  const float* Kp0 = K + (size_t)h * SK * DK
                       + (((size_t)wave * 16 + m) * DK) + 2 * hh;

  // Software pipeline: double-buffered K-tile registers so the 16-load clause for
  // tile i+1 is in flight while the 16-WMMA chain + exp update for tile i executes.
  v2f ka[16], kb[16];
  load_ktile(ka, Kp0);
  for (int i = 0; i < KT_PER_WAVE; i += 2) {
    // prefetch tile i+1 (always in range: i+1 <= 31)
    load_ktile(kb, Kp0 + (size_t)(i + 1) * KT_STRIDE);
    consume_tile(qa, ka, accS, accM, accZ, accS2);
    // prefetch tile i+2 (out of range only on the last iteration)
    if (i + 2 < KT_PER_WAVE)
      load_ktile(ka, Kp0 + (size_t)(i + 2) * KT_STRIDE);
    consume_tile(qa, kb, accS, accM, accZ, accS2);
  }

#pragma unroll
  for (int j = 0; j < 8; ++j) {
    accS[j]  = xor16_sum(accS[j]);
    accM[j]  = xor16_max(accM[j]);
    accZ[j]  = xor16_sum(accZ[j]);
    accS2[j] = xor16_sum(accS2[j]);
  }
  if (m == 0) {
    const int rbase = hh * 8;   // lanes 0..15 -> rows 0..7, lanes 16..31 -> rows 8..15
#pragma unroll
    for (int j = 0; j < 8; ++j) {
      wstats[wave][rbase + j][0] = accS[j];
      wstats[wave][rbase + j][1] = accM[j];
      wstats[wave][rbase + j][2] = accZ[j];
      wstats[wave][rbase + j][3] = accS2[j];
    }
  }
  __syncthreads();

  if (tid < 64) {               // combine 8 waves: thread = (row, stat)
    const int row = tid >> 2, st = tid & 3;
    float v = wstats[0][row][st];
    for (int w = 1; w < WAVES; ++w) {
      float u = wstats[w][row][st];
      v = (st == 1) ? fmaxf(v, u) : (v + u);
    }
    wstats[0][row][st] = v;
  }
  __syncthreads();

  if (tid < 16) {
    const float Srow  = wstats[0][tid][0];
    const float Mrow  = wstats[0][tid][1];
    const float Zrow  = wstats[0][tid][2];
    const float S2row = wstats[0][tid][3];
    const float p2  = S2row / (Zrow * Zrow);                       // sum of probs^2
    const float var = (p2 - (1.0f / (float)SK)) * (1.0f / ((float)SK - 1.0f));
    rowtrip[tid][0] = Srow;
    rowtrip[tid][1] = Mrow;
    rowtrip[tid][2] = var;
  }
  __syncthreads();

  if (tid == 0) {
    float a = 0.f, b = 0.f, cv = 0.f;
    for (int r = 0; r < 16; ++r) {
      a += rowtrip[r][0]; b += rowtrip[r][1]; cv += rowtrip[r][2];
    }
    partials[(size_t)bid * 3 + 0] = a;
    partials[(size_t)bid * 3 + 1] = b;
    partials[(size_t)bid * 3 + 2] = cv;
  }
}

// One block per head: deterministic tree-sum of 256 q-tile partials, then the MLP.
__global__ __launch_bounds__(256)
void finalize_kernel(const float* __restrict__ partials,
                     const float* __restrict__ W1, const float* __restrict__ b1,
                     const float* __restrict__ W2, const float* __restrict__ b2,
                     float* __restrict__ out) {
  const int h = blockIdx.x;
  const int tid = threadIdx.x;
  __shared__ float red[256][3];
  const float* p = partials + (((size_t)h * QTILES) + tid) * 3;
  red[tid][0] = p[0]; red[tid][1] = p[1]; red[tid][2] = p[2];
  __syncthreads();
  for (int s = 128; s > 0; s >>= 1) {
    if (tid < s) {
      red[tid][0] += red[tid + s][0];
      red[tid][1] += red[tid + s][1];
      red[tid][2] += red[tid + s][2];
    }
    __syncthreads();
  }
  if (tid == 0) {
    float mean_sim = red[0][0] * (1.0f / ((float)SQ * (float)SK));
    float max_sim  = red[0][1] * (1.0f / (float)SQ);
    float entropy  = red[0][2] * (1.0f / (float)SQ);
    mean_sim = fminf(fmaxf(mean_sim, -10.f), 10.f);
    max_sim  = fminf(fmaxf(max_sim,  -10.f), 10.f);
    entropy  = fminf(fmaxf(entropy,    0.f),  1.f);
    float racc = b2[0];
#pragma unroll
    for (int i = 0; i < 16; ++i) {
      float hs = b1[i] + W1[i * 3 + 0] * mean_sim
                       + W1[i * 3 + 1] * max_sim
                       + W1[i * 3 + 2] * entropy;
      racc += W2[i] * tanhf(hs);
    }
    float t = 0.1f + 0.9f / (1.0f + __expf(-racc));
    t = fminf(fmaxf(t, 0.1f), 1.0f);
    out[h] = t;
  }
}

extern "C" void kernel_launch(void* const* d_in, const int* in_sizes, int n_in,
                              void* d_out, int out_size, void* d_ws, size_t ws_size,
                              hipStream_t stream) {
  const float* Q  = (const float*)d_in[0];
  const float* K  = (const float*)d_in[1];
  const float* W1 = (const float*)d_in[2];
  const float* b1 = (const float*)d_in[3];
  const float* W2 = (const float*)d_in[4];
  const float* b2 = (const float*)d_in[5];
  float* out = (float*)d_out;
  float* partials = (float*)d_ws;   // 16*256*3 floats = 48 KB, fully overwritten each call

  attn_stats_kernel<<<NHEAD * QTILES, 256, 0, stream>>>(Q, K, partials);
  finalize_kernel<<<NHEAD, 256, 0, stream>>>(partials, W1, b1, W2, b2, out);
}